// GCNIILayer_35802847380157
// MI455X (gfx1250) — compile-verified
//
#include <hip/hip_runtime.h>

#define N_NODES 100000
#define E_EDGES 3200000
#define DIN     128
#define DOUT    64
#define ALPHA   0.1f
// theta*support + (1-theta)*support == support, so theta drops out entirely.

typedef __attribute__((ext_vector_type(2))) float v2f;
typedef __attribute__((ext_vector_type(8))) float v8f;

// ---------------------------------------------------------------------------
// Kernel 1: out = ALPHA * h0 + bias   (float4 vectorized; N*DOUT = 6.4M f32)
// ---------------------------------------------------------------------------
__global__ void gcnii_init_kernel(const float* __restrict__ h0,
                                  const float* __restrict__ bias,
                                  float* __restrict__ out) {
    const int idx = blockIdx.x * blockDim.x + threadIdx.x;   // float4 index
    const int total4 = (N_NODES * DOUT) / 4;
    if (idx >= total4) return;
    const float4 h = ((const float4*)h0)[idx];
    const float4 b = ((const float4*)bias)[idx & (DOUT / 4 - 1)]; // feature group
    float4 o;
    o.x = ALPHA * h.x + b.x;
    o.y = ALPHA * h.y + b.y;
    o.z = ALPHA * h.z + b.z;
    o.w = ALPHA * h.w + b.w;
    ((float4*)out)[idx] = o;
}

// ---------------------------------------------------------------------------
// Kernel 2: xw = x @ W via V_WMMA_F32_16X16X4_F32.
// Block = 256 threads = 8 waves; each wave computes a 16(M) x 64(N) tile.
// Weight (128x64 f32 = 32KB) staged through LDS once per block.
//
// Fragment layouts per ISA 7.12.2 (wave32):
//   A 16x4 f32 : lanes 0-15 -> M=lane, VGPR{0,1}=K{0,1}; lanes 16-31 -> K{2,3}
//   B 4x16 f32 : lanes 0-15 -> N=lane, VGPR{0,1}=K{0,1}; lanes 16-31 -> K{2,3}
//   C/D 16x16  : VGPR i -> M=i (lanes 0-15), M=i+8 (lanes 16-31), N = lane&15
// ---------------------------------------------------------------------------
__global__ void __launch_bounds__(256)
gcnii_gemm_wmma_kernel(const float* __restrict__ x,
                       const float* __restrict__ w,   // [DIN][DOUT]
                       float* __restrict__ xw) {      // [N][DOUT]
    __shared__ float sw[DIN * DOUT];                  // 32 KB

    // Cooperative weight load: 256 threads x 8 float4 = 2048 float4.
    {
        const float4* wsrc = (const float4*)w;
        float4*       wdst = (float4*)sw;
        #pragma unroll
        for (int i = 0; i < (DIN * DOUT / 4) / 256; ++i)
            wdst[threadIdx.x + i * 256] = wsrc[threadIdx.x + i * 256];
    }
    __syncthreads();

    const int waveInBlk = threadIdx.x >> 5;
    const int lane      = threadIdx.x & 31;
    const int m         = lane & 15;                  // sub-lane row/col id
    const bool lo       = (lane < 16);

    const int tile = blockIdx.x * 8 + waveInBlk;      // 16-row tile index
    if (tile >= N_NODES / 16) return;                 // 6250 tiles, exact fit
    const int rowBase = tile * 16;

    v8f acc0 = {}, acc1 = {}, acc2 = {}, acc3 = {};

    const float* xrow = x + (size_t)(rowBase + m) * DIN;

    #pragma unroll 4
    for (int kk = 0; kk < DIN; kk += 4) {
        // A fragment: this lane's half-K pair
        const int kA = kk + (lo ? 0 : 2);
        v2f a;
        a.x = xrow[kA + 0];
        a.y = xrow[kA + 1];

        // B fragments for the four 16-wide N tiles
        const int kB0 = kk + (lo ? 0 : 2);
        const int kB1 = kk + (lo ? 1 : 3);
        v2f b0, b1, b2, b3;
        b0.x = sw[kB0 * DOUT + m +  0]; b0.y = sw[kB1 * DOUT + m +  0];
        b1.x = sw[kB0 * DOUT + m + 16]; b1.y = sw[kB1 * DOUT + m + 16];
        b2.x = sw[kB0 * DOUT + m + 32]; b2.y = sw[kB1 * DOUT + m + 32];
        b3.x = sw[kB0 * DOUT + m + 48]; b3.y = sw[kB1 * DOUT + m + 48];

        acc0 = __builtin_amdgcn_wmma_f32_16x16x4_f32(false, a, false, b0,
                                                     (short)0, acc0, false, false);
        acc1 = __builtin_amdgcn_wmma_f32_16x16x4_f32(false, a, false, b1,
                                                     (short)0, acc1, false, false);
        acc2 = __builtin_amdgcn_wmma_f32_16x16x4_f32(false, a, false, b2,
                                                     (short)0, acc2, false, false);
        acc3 = __builtin_amdgcn_wmma_f32_16x16x4_f32(false, a, false, b3,
                                                     (short)0, acc3, false, false);
    }

    // Store D tiles: VGPR i holds M = i (lanes 0-15) / i+8 (lanes 16-31).
    #pragma unroll
    for (int i = 0; i < 8; ++i) {
        const int r = rowBase + (lo ? i : i + 8);
        float* orow = xw + (size_t)r * DOUT + m;
        orow[ 0] = acc0[i];
        orow[16] = acc1[i];
        orow[32] = acc2[i];
        orow[48] = acc3[i];
    }
}

// ---------------------------------------------------------------------------
// Kernel 3: segment-sum SpMM with sorted rows.
// One wave owns a contiguous chunk of edges; each lane carries 2 of the 64
// features in registers. Flush with global atomics only on row-run boundaries
// (avg run length = 32 edges) and at chunk edges.
//   out[r] += (1-ALPHA) * sum_e vals[e] * xw[cols[e]]
// ---------------------------------------------------------------------------
#define EDGES_PER_WAVE 512

__global__ void __launch_bounds__(256)
gcnii_spmm_kernel(const int* __restrict__ rows,
                  const int* __restrict__ cols,
                  const float* __restrict__ vals,
                  const float* __restrict__ xw,
                  float* __restrict__ out) {
    const int wave = (blockIdx.x * blockDim.x + threadIdx.x) >> 5;
    const int lane = threadIdx.x & 31;

    long start = (long)wave * EDGES_PER_WAVE;
    if (start >= E_EDGES) return;
    long end = start + EDGES_PER_WAVE;
    if (end > E_EDGES) end = E_EDGES;

    float acc0 = 0.0f, acc1 = 0.0f;
    int cur = rows[start];                 // wave-uniform (all lanes same addr)

    for (long e = start; e < end; ++e) {
        const int r = rows[e];
        if (r != cur) {
            atomicAdd(&out[(size_t)cur * DOUT + lane],      acc0);
            atomicAdd(&out[(size_t)cur * DOUT + lane + 32], acc1);
            acc0 = 0.0f; acc1 = 0.0f;
            cur = r;
        }
        const int   c = cols[e];
        const float v = (1.0f - ALPHA) * vals[e];
        const float* src = xw + (size_t)c * DOUT;
        acc0 += v * src[lane];
        acc1 += v * src[lane + 32];
    }
    atomicAdd(&out[(size_t)cur * DOUT + lane],      acc0);
    atomicAdd(&out[(size_t)cur * DOUT + lane + 32], acc1);
}

// ---------------------------------------------------------------------------
// Launch
// ---------------------------------------------------------------------------
extern "C" void kernel_launch(void* const* d_in, const int* in_sizes, int n_in,
                              void* d_out, int out_size, void* d_ws, size_t ws_size,
                              hipStream_t stream) {
    const float* x        = (const float*)d_in[0];
    const int*   adj_rows = (const int*)  d_in[1];
    const int*   adj_cols = (const int*)  d_in[2];
    const float* adj_vals = (const float*)d_in[3];
    const float* h0       = (const float*)d_in[4];
    const float* weight   = (const float*)d_in[5];
    const float* bias     = (const float*)d_in[6];
    float*       out      = (float*)d_out;
    float*       xw       = (float*)d_ws;            // N*DOUT f32 = 25.6 MB

    // 1) out = ALPHA*h0 + bias
    {
        const int total4 = (N_NODES * DOUT) / 4;
        const int blocks = (total4 + 255) / 256;
        gcnii_init_kernel<<<blocks, 256, 0, stream>>>(h0, bias, out);
    }

    // 2) xw = x @ W (WMMA f32)
    {
        const int tiles  = N_NODES / 16;             // 6250
        const int blocks = (tiles + 7) / 8;          // 8 waves/block
        gcnii_gemm_wmma_kernel<<<blocks, 256, 0, stream>>>(x, weight, xw);
    }

    // 3) out += (1-ALPHA) * SpMM(adj, xw)
    {
        const int waves  = (E_EDGES + EDGES_PER_WAVE - 1) / EDGES_PER_WAVE;
        const int blocks = (waves + 7) / 8;          // 8 waves/block
        gcnii_spmm_kernel<<<blocks, 256, 0, stream>>>(adj_rows, adj_cols,
                                                      adj_vals, xw, out);
    }
}